// SE3TrAttention_5231270166739
// MI455X (gfx1250) — compile-verified
//
#include <hip/hip_runtime.h>
#include <hip/hip_bf16.h>

// ---------------- problem constants (match reference) ----------------
constexpr int E_EDGES = 32768;
constexpr int NNODES  = 4096;
constexpr int EMB     = 64;
constexpr int M_IN    = 4;     // (L_IN+1)^2
constexpr int C_IN    = 16;
constexpr int C_OUT   = 32;
constexpr int H       = 4;
constexpr int NW      = 6 * C_IN * C_OUT;   // 3072 flat W_k columns (w,c,o)
constexpr int TE      = 16;                 // edges per tile (WMMA M)
constexpr int NTILES  = NW / 16;            // 192 column tiles of width 16

typedef float v2f __attribute__((ext_vector_type(2)));
typedef float v8f __attribute__((ext_vector_type(8)));

#if __has_builtin(__builtin_amdgcn_wmma_f32_16x16x4_f32)
#define HAVE_WMMA_F32X4 1
#else
#define HAVE_WMMA_F32X4 0
#endif

// =====================================================================
// Kernel 0: repack W_p (64 x 3072, row major) into WMMA B-fragment order:
//   Wt[(r>>1)*2*NW + 2*n + (r&1)] = W_p[r*NW + n]
// so a lane's (k, k+1) B pair is one contiguous 8-byte global_load_b64.
// =====================================================================
__global__ void pack_wp_kernel(const float* __restrict__ W_p,
                               float* __restrict__ Wt) {
  int p = blockIdx.x * blockDim.x + threadIdx.x;    // output index
  if (p >= EMB * NW) return;
  int j  = p & 1;
  int q  = p >> 1;
  int n  = q % NW;
  int rh = q / NW;
  Wt[p] = W_p[(size_t)(rh * 2 + j) * NW + n];
}

// =====================================================================
// Kernel 1: fused  q  +  (emb @ W_p) GEMM  +  z = <V, W_k>  per 16-edge tile
// One workgroup (256 thr = 8 waves) per 16 edges.
// Wave w handles n-tile pairs {w, w+8}, {w+16, w+24}, ...  (stride 16).
// All of a wave's tiles satisfy nt&1 == wave&1, so its lanes map to a
// FIXED head pair -> single zacc[8]. Two independent WMMA accumulation
// chains per iteration hide the XDL latency.
// =====================================================================
__launch_bounds__(256)
__global__ void fused_z_kernel(const float* __restrict__ x_q,
                               const float* __restrict__ x_k,
                               const float* __restrict__ emb,
                               const float* __restrict__ W_q,
                               const float* __restrict__ Wt,
                               float* __restrict__ z_out) {
  __shared__ float s_emb[TE][EMB];          // 4 KB  (A matrix: 16 edges x 64)
  __shared__ float s_xq [TE][M_IN][C_IN];   // 4 KB
  __shared__ float s_xk [TE][M_IN][C_IN];   // 4 KB
  __shared__ float s_q  [TE][M_IN][C_OUT];  // 8 KB
  __shared__ float s_z  [TE][H];            // 256 B

  const int tid = threadIdx.x;
  const int eb  = blockIdx.x * TE;

  // ---- phase 0: stage tiles into LDS ----
  for (int i = tid; i < TE * EMB; i += 256)
    s_emb[i >> 6][i & 63] = emb[(size_t)(eb + (i >> 6)) * EMB + (i & 63)];
  for (int i = tid; i < TE * M_IN * C_IN; i += 256) {
    int e = i >> 6, r = i & 63;
    s_xq[e][r >> 4][r & 15] = x_q[(size_t)(eb + e) * (M_IN * C_IN) + r];
    s_xk[e][r >> 4][r & 15] = x_k[(size_t)(eb + e) * (M_IN * C_IN) + r];
  }
  if (tid < TE * H) s_z[tid >> 2][tid & 3] = 0.f;
  __syncthreads();

  // ---- phase 1: q tile (shared W_q; so2 structure unrolled for L=1) ----
  // Each thread has fixed (m,o) across its 8 edges -> W_q loads are hoisted.
  for (int i = tid; i < TE * M_IN * C_OUT; i += 256) {
    int e = i >> 7, m = (i >> 5) & 3, o = i & 31;
    float acc = 0.f;
    if (m == 0) {        // out0 = x0@W0 + x2@W2
      #pragma unroll
      for (int c = 0; c < C_IN; ++c)
        acc += s_xq[e][0][c] * W_q[(0 * C_IN + c) * C_OUT + o]
             + s_xq[e][2][c] * W_q[(2 * C_IN + c) * C_OUT + o];
    } else if (m == 2) { // out2 = x0@W1 + x2@W3
      #pragma unroll
      for (int c = 0; c < C_IN; ++c)
        acc += s_xq[e][0][c] * W_q[(1 * C_IN + c) * C_OUT + o]
             + s_xq[e][2][c] * W_q[(3 * C_IN + c) * C_OUT + o];
    } else if (m == 3) { // out3 = x3@W4 - x1@W5
      #pragma unroll
      for (int c = 0; c < C_IN; ++c)
        acc += s_xq[e][3][c] * W_q[(4 * C_IN + c) * C_OUT + o]
             - s_xq[e][1][c] * W_q[(5 * C_IN + c) * C_OUT + o];
    } else {             // out1 = x3@W5 + x1@W4
      #pragma unroll
      for (int c = 0; c < C_IN; ++c)
        acc += s_xq[e][3][c] * W_q[(5 * C_IN + c) * C_OUT + o]
             + s_xq[e][1][c] * W_q[(4 * C_IN + c) * C_OUT + o];
    }
    s_q[e][m][o] = acc;
  }
  __syncthreads();

  // ---- phase 2: WMMA GEMM over W_k columns, fused <V, W_k> reduction ----
  const int wave = tid >> 5;
  const int lane = tid & 31;
  const int lrow = lane & 15;   // A-matrix M row / B,D column within tile
  const int lhi  = lane >> 4;   // lane half

  float zacc[8];
  #pragma unroll
  for (int v = 0; v < 8; ++v) zacc[v] = 0.f;

  // V-weighting of one finished accumulator tile (widx/c are wave-uniform:
  // a 16-wide tile never crosses a 512-wide w-block -> force scalar).
  auto consume = [&](int nbase, const v8f& acc) {
    const int widx = __builtin_amdgcn_readfirstlane(nbase >> 9);        // 0..5
    const int c    = __builtin_amdgcn_readfirstlane((nbase >> 5) & 15); // 0..15
    const int o    = (nbase & 16) | lrow;                               // lane-varying
    if (widx < 4) {
      const int i0 = widx & 2;          // x_k row
      const int m0 = (widx & 1) * 2;    // q row
      #pragma unroll
      for (int v = 0; v < 8; ++v) {
        const int e = v + 8 * lhi;
        zacc[v] += (s_xk[e][i0][c] * s_q[e][m0][o]) * acc[v];
      }
    } else if (widx == 4) {             // V4 = x3*q3 + x1*q1
      #pragma unroll
      for (int v = 0; v < 8; ++v) {
        const int e = v + 8 * lhi;
        float V = s_xk[e][3][c] * s_q[e][3][o] + s_xk[e][1][c] * s_q[e][1][o];
        zacc[v] += V * acc[v];
      }
    } else {                            // V5 = x3*q1 - x1*q3
      #pragma unroll
      for (int v = 0; v < 8; ++v) {
        const int e = v + 8 * lhi;
        float V = s_xk[e][3][c] * s_q[e][1][o] - s_xk[e][1][c] * s_q[e][3][o];
        zacc[v] += V * acc[v];
      }
    }
  };

#if HAVE_WMMA_F32X4
  // A fragments are nt-invariant: hoist all 16 k-step fragments.
  v2f afrag[EMB / 4];
  #pragma unroll
  for (int ks = 0; ks < EMB / 4; ++ks) {
    afrag[ks][0] = s_emb[lrow][4 * ks + 2 * lhi + 0];
    afrag[ks][1] = s_emb[lrow][4 * ks + 2 * lhi + 1];
  }
  // lane-fixed part of the packed-B address
  const float* wt_lane = Wt + (size_t)lhi * (2 * NW);

  for (int nt = wave; nt < NTILES; nt += 16) {     // 12 iters, 2 tiles each
    const int nb0 = nt * 16;
    const int nb1 = nb0 + 128;                     // tile nt+8 (same parity)
    const float* bp0 = wt_lane + 2 * (nb0 + lrow);
    const float* bp1 = wt_lane + 2 * (nb1 + lrow);
    v8f acc0 = {}, acc1 = {};
    #pragma unroll
    for (int ks = 0; ks < EMB / 4; ++ks) {
      const size_t koff = (size_t)ks * (4 * NW);   // 2 packed rows per k-step
      v2f b0 = *(const v2f*)(bp0 + koff);          // global_load_b64
      v2f b1 = *(const v2f*)(bp1 + koff);
      acc0 = __builtin_amdgcn_wmma_f32_16x16x4_f32(
          false, afrag[ks], false, b0, (short)0, acc0, false, false);
      acc1 = __builtin_amdgcn_wmma_f32_16x16x4_f32(
          false, afrag[ks], false, b1, (short)0, acc1, false, false);
    }
    consume(nb0, acc0);
    consume(nb1, acc1);
  }
#else
  for (int nt = wave; nt < NTILES; nt += 8) {
    const int nbase = nt * 16;
    v8f acc = {};
    for (int d = 0; d < EMB; ++d) {
      // Wt[(d>>1)*2*NW + 2*n + (d&1)]
      float bv = Wt[(size_t)(d >> 1) * (2 * NW) + 2 * (nbase + lrow) + (d & 1)];
      #pragma unroll
      for (int v = 0; v < 8; ++v) acc[v] += s_emb[v + 8 * lhi][d] * bv;
    }
    consume(nbase, acc);
  }
#endif

  // ---- phase 3: reduce per-lane partials into z tile, write out ----
  // nt === wave (mod 2) -> this wave's lanes always hit head pair (wave&1).
  const int h = ((wave & 1) << 1) | (lrow >> 3);
  #pragma unroll
  for (int v = 0; v < 8; ++v)
    atomicAdd(&s_z[v + 8 * lhi][h], zacc[v]);     // ds_add_f32
  __syncthreads();
  if (tid < TE * H) {
    constexpr float scale = 0.70710678118654752440f;  // (K_CH/H)^-0.5
    z_out[(size_t)(eb + (tid >> 2)) * H + (tid & 3)] =
        s_z[tid >> 2][tid & 3] * scale;
  }
}

// =====================================================================
// Kernel 2: per-(node, head) online segment max + sum(exp) — deterministic
// (index is sorted, so each node's edges are a contiguous range).
// =====================================================================
__global__ void node_stats_kernel(const float* __restrict__ z,
                                  const int* __restrict__ index,
                                  float* __restrict__ nmax,
                                  float* __restrict__ nsum) {
  int t = blockIdx.x * blockDim.x + threadIdx.x;
  if (t >= NNODES * H) return;
  int node = t >> 2, h = t & 3;

  int lo = 0, hi = E_EDGES;
  while (lo < hi) { int mid = (lo + hi) >> 1; if (index[mid] < node) lo = mid + 1; else hi = mid; }
  int start = lo;
  hi = E_EDGES;
  while (lo < hi) { int mid = (lo + hi) >> 1; if (index[mid] < node + 1) lo = mid + 1; else hi = mid; }
  int end = lo;

  float m = -__builtin_inff(), s = 0.f;
  for (int e = start; e < end; ++e) {
    float v = z[(size_t)e * H + h];
    if (v > m) { s = s * __expf(m - v) + 1.f; m = v; }
    else       { s += __expf(v - m); }
  }
  nmax[t] = m;
  nsum[t] = s;
}

// =====================================================================
// Kernel 3: out = exp(z - max[index]) / sum[index]
// =====================================================================
__global__ void softmax_out_kernel(const float* __restrict__ z,
                                   const int* __restrict__ index,
                                   const float* __restrict__ nmax,
                                   const float* __restrict__ nsum,
                                   float* __restrict__ out) {
  int t = blockIdx.x * blockDim.x + threadIdx.x;
  if (t >= E_EDGES * H) return;
  int e = t >> 2, h = t & 3;
  int n = index[e];
  out[t] = __expf(z[t] - nmax[n * H + h]) / nsum[n * H + h];
}

// =====================================================================
extern "C" void kernel_launch(void* const* d_in, const int* in_sizes, int n_in,
                              void* d_out, int out_size, void* d_ws, size_t ws_size,
                              hipStream_t stream) {
  const float* x_q   = (const float*)d_in[0];
  const float* x_k   = (const float*)d_in[1];
  const float* emb   = (const float*)d_in[2];
  const int*   index = (const int*)  d_in[3];
  const float* W_q   = (const float*)d_in[4];
  const float* W_p   = (const float*)d_in[5];
  // d_in[6] = num_nodes scalar (compile-time constant here)

  float* z    = (float*)d_ws;                    // E*H floats      (512 KB)
  float* nmax = z + (size_t)E_EDGES * H;         // NNODES*H        ( 64 KB)
  float* nsum = nmax + (size_t)NNODES * H;       // NNODES*H        ( 64 KB)
  float* Wt   = nsum + (size_t)NNODES * H;       // EMB*NW          (768 KB)

  pack_wp_kernel<<<(EMB * NW + 255) / 256, 256, 0, stream>>>(W_p, Wt);
  fused_z_kernel<<<E_EDGES / TE, 256, 0, stream>>>(x_q, x_k, emb, W_q, Wt, z);
  node_stats_kernel<<<(NNODES * H + 255) / 256, 256, 0, stream>>>(z, index, nmax, nsum);
  softmax_out_kernel<<<(E_EDGES * H + 255) / 256, 256, 0, stream>>>(z, index, nmax, nsum,
                                                                    (float*)d_out);
}